// PdnBLOCK_68831145886181
// MI455X (gfx1250) — compile-verified
//
#include <hip/hip_runtime.h>
#include <hip/hip_bf16.h>
#include <stdint.h>

#define N_NODES   50000
#define N_EDGES   800000
#define PE        64
#define PD        16
#define N_GRAPHS  256
#define EPS       1e-5f
#define NEG_SLOPE 0.01f

typedef __attribute__((ext_vector_type(2))) float v2f;
typedef __attribute__((ext_vector_type(8))) float v8f;

// ---------------------------------------------------------------------------
// Edge kernel: one wave = one 16-edge tile.
// msg[16x64] = x[src] @ W_msg + attr @ W_edge + (b_msg + b_edge)
// computed with V_WMMA_F32_16X16X4_F32, then atomic scatter-add into agg[dst].
// agg is pre-initialized with x (skip connection).
// ---------------------------------------------------------------------------
__global__ __launch_bounds__(256) void edge_msg_kernel(
    const float* __restrict__ x,
    const int*   __restrict__ src,
    const int*   __restrict__ dst,
    const float* __restrict__ attr,
    const float* __restrict__ Wm,   // [64][64] (k-major rows)
    const float* __restrict__ bm,   // [64]
    const float* __restrict__ We,   // [16][64]
    const float* __restrict__ be,   // [64]
    float*       __restrict__ agg)
{
    __shared__ float sWm[PE * PE];       // 16 KB
    __shared__ float sWe[PD * PE];       //  4 KB
    __shared__ float sX[8][16 * PE];     // 32 KB (per-wave gathered x rows)
    __shared__ float sA[8][16 * PD];     //  8 KB (per-wave attr rows)
    __shared__ int   sD[8][16];

    const int tid  = threadIdx.x;
    const int wave = tid >> 5;
    const int lane = tid & 31;

    // Stage weights block-wide (vectorized, fully L2-resident).
    {
        const float4* Wm4  = (const float4*)Wm;
        float4*       sWm4 = (float4*)sWm;
        #pragma unroll
        for (int i = tid; i < PE * PE / 4; i += 256) sWm4[i] = Wm4[i];
        const float4* We4  = (const float4*)We;
        float4*       sWe4 = (float4*)sWe;
        if (tid < PD * PE / 4) sWe4[tid] = We4[tid];
    }

    const int tile = blockIdx.x * 8 + wave;
    const int e0   = tile * 16;

    if (lane < 16) sD[wave][lane] = dst[e0 + lane];

    // Gather: each lane stages half of one x-row (32 floats) + half an attr row.
    {
        const int row = lane >> 1;
        const int hlf = lane & 1;
        const int s   = src[e0 + row];
        const float4* xr = (const float4*)(x + (size_t)s * PE + hlf * 32);
        float4*       ox = (float4*)(&sX[wave][row * PE + hlf * 32]);
        #pragma unroll
        for (int i = 0; i < 8; ++i) ox[i] = xr[i];
        const float4* ar = (const float4*)(attr + (size_t)(e0 + row) * PD + hlf * 8);
        float4*       oa = (float4*)(&sA[wave][row * PD + hlf * 8]);
        oa[0] = ar[0]; oa[1] = ar[1];
    }
    __syncthreads();

    // WMMA fragment coordinates (wave32, ISA 7.12.2 layouts):
    // A (16x4 f32):  M = lane&15,  VGPR{0,1} hold K = k0+koff, k0+koff+1
    // B (4x16 f32):  N = lane&15,  mirrored K split by lane half
    // C/D (16x16):   VGPR r -> M = r + 8*(lane>>4), N = lane&15
    const int col  = lane & 15;
    const int hi   = lane >> 4;    // 0 or 1
    const int koff = hi * 2;

    #pragma unroll
    for (int n = 0; n < 4; ++n) {
        const int n0 = n * 16;
        const float bias = bm[n0 + col] + be[n0 + col];
        v8f acc;
        #pragma unroll
        for (int r = 0; r < 8; ++r) acc[r] = bias;

        // x_src[16x64] @ W_msg[64x64] tile: 16 K-steps of 16x16x4
        #pragma unroll
        for (int kk = 0; kk < 16; ++kk) {
            const int k0 = kk * 4;
            v2f a = *(const v2f*)(&sX[wave][col * PE + k0 + koff]);
            v2f b;
            b[0] = sWm[(k0 + koff    ) * PE + n0 + col];
            b[1] = sWm[(k0 + koff + 1) * PE + n0 + col];
            acc = __builtin_amdgcn_wmma_f32_16x16x4_f32(
                false, a, false, b, (short)0, acc, false, false);
        }
        // attr[16x16] @ W_edge[16x64] tile: 4 K-steps
        #pragma unroll
        for (int kk = 0; kk < 4; ++kk) {
            const int k0 = kk * 4;
            v2f a = *(const v2f*)(&sA[wave][col * PD + k0 + koff]);
            v2f b;
            b[0] = sWe[(k0 + koff    ) * PE + n0 + col];
            b[1] = sWe[(k0 + koff + 1) * PE + n0 + col];
            acc = __builtin_amdgcn_wmma_f32_16x16x4_f32(
                false, a, false, b, (short)0, acc, false, false);
        }
        // Scatter-add (L2 atomics; agg fits in L2).
        #pragma unroll
        for (int r = 0; r < 8; ++r) {
            const int m = r + hi * 8;
            const int d = sD[wave][m];
            atomicAdd(&agg[(size_t)d * PE + n0 + col], acc[r]);
        }
    }
}

// ---------------------------------------------------------------------------
// GraphNorm pipeline (per-graph stats live in 64 KB buffers -> pure L2 traffic)
// ---------------------------------------------------------------------------
__global__ void count_kernel(const int* __restrict__ batch, float* __restrict__ counts) {
    int i = blockIdx.x * blockDim.x + threadIdx.x;
    if (i < N_NODES) atomicAdd(&counts[batch[i]], 1.0f);
}

__global__ void mean_accum_kernel(const float* __restrict__ h,
                                  const int* __restrict__ batch,
                                  float* __restrict__ mean) {
    int i = blockIdx.x * blockDim.x + threadIdx.x;   // over N_NODES*PE
    int n = i >> 6, f = i & 63;
    atomicAdd(&mean[batch[n] * PE + f], h[i]);
}

__global__ void mean_fin_kernel(float* __restrict__ mean, const float* __restrict__ counts) {
    int i = blockIdx.x * blockDim.x + threadIdx.x;   // N_GRAPHS*PE
    mean[i] /= fmaxf(counts[i >> 6], 1.0f);
}

__global__ void center_var_kernel(const float* __restrict__ h,
                                  const int* __restrict__ batch,
                                  const float* __restrict__ mean,
                                  const float* __restrict__ mscale,
                                  float* __restrict__ tmp,
                                  float* __restrict__ var) {
    int i = blockIdx.x * blockDim.x + threadIdx.x;
    int n = i >> 6, f = i & 63;
    int g = batch[n];
    float o = h[i] - mean[g * PE + f] * mscale[f];
    tmp[i] = o;
    atomicAdd(&var[g * PE + f], o * o);
}

__global__ void scale_fin_kernel(float* __restrict__ var,
                                 const float* __restrict__ counts,
                                 const float* __restrict__ w) {
    int i = blockIdx.x * blockDim.x + threadIdx.x;   // N_GRAPHS*PE
    float c = fmaxf(counts[i >> 6], 1.0f);
    var[i] = w[i & 63] * rsqrtf(var[i] / c + EPS);   // weight / std
}

__global__ void apply_kernel(const float* __restrict__ tmp,
                             const int* __restrict__ batch,
                             const float* __restrict__ scale,
                             const float* __restrict__ bias,
                             float* __restrict__ out) {
    int i = blockIdx.x * blockDim.x + threadIdx.x;
    int n = i >> 6, f = i & 63;
    float v = tmp[i] * scale[batch[n] * PE + f] + bias[f];
    out[i] = v > 0.0f ? v : NEG_SLOPE * v;
}

// ---------------------------------------------------------------------------
extern "C" void kernel_launch(void* const* d_in, const int* in_sizes, int n_in,
                              void* d_out, int out_size, void* d_ws, size_t ws_size,
                              hipStream_t stream) {
    const float* x0    = (const float*)d_in[0];
    const int*   idx   = (const int*)  d_in[1];
    const float* attr  = (const float*)d_in[2];
    const int*   batch = (const int*)  d_in[3];
    const float* Wm    = (const float*)d_in[4];
    const float* bm    = (const float*)d_in[5];
    const float* We    = (const float*)d_in[6];
    const float* be    = (const float*)d_in[7];
    const float* gw    = (const float*)d_in[8];
    const float* gb    = (const float*)d_in[9];
    const float* gms   = (const float*)d_in[10];

    const int* src = idx;
    const int* dst = idx + N_EDGES;

    char* ws = (char*)d_ws;
    const size_t node_bytes = (size_t)N_NODES * PE * sizeof(float);
    float* h      = (float*)(ws);
    float* tmp    = (float*)(ws + node_bytes);
    float* xbuf   = (float*)(ws + 2 * node_bytes);
    float* mean   = (float*)(ws + 3 * node_bytes);
    float* var    = mean + N_GRAPHS * PE;
    float* counts = var  + N_GRAPHS * PE;

    // Per-graph node counts (computed once, reused by both layers).
    hipMemsetAsync(counts, 0, N_GRAPHS * sizeof(float), stream);
    count_kernel<<<(N_NODES + 255) / 256, 256, 0, stream>>>(batch, counts);

    const int NE = N_NODES * PE;     // 3.2M, divisible by 256
    for (int l = 0; l < 2; ++l) {
        const float* xin  = (l == 0) ? x0 : xbuf;
        float*       xout = (l == 1) ? (float*)d_out : xbuf;

        // agg := x  (skip connection baked into the scatter target)
        hipMemcpyAsync(h, xin, node_bytes, hipMemcpyDeviceToDevice, stream);

        edge_msg_kernel<<<N_EDGES / 16 / 8, 256, 0, stream>>>(
            xin, src, dst, attr,
            Wm + (size_t)l * PE * PE, bm + (size_t)l * PE,
            We + (size_t)l * PD * PE, be + (size_t)l * PE, h);

        hipMemsetAsync(mean, 0, (size_t)2 * N_GRAPHS * PE * sizeof(float), stream); // mean+var
        mean_accum_kernel<<<NE / 256, 256, 0, stream>>>(h, batch, mean);
        mean_fin_kernel<<<N_GRAPHS * PE / 256, 256, 0, stream>>>(mean, counts);
        center_var_kernel<<<NE / 256, 256, 0, stream>>>(h, batch, mean,
                                                        gms + (size_t)l * PE, tmp, var);
        scale_fin_kernel<<<N_GRAPHS * PE / 256, 256, 0, stream>>>(var, counts,
                                                                  gw + (size_t)l * PE);
        apply_kernel<<<NE / 256, 256, 0, stream>>>(tmp, batch, var,
                                                   gb + (size_t)l * PE, xout);
    }
}